// JKNet_12695923327568
// MI455X (gfx1250) — compile-verified
//
#include <hip/hip_runtime.h>

// ---------------- problem constants (match reference) ----------------
#define N_NODES  50000
#define N_EDGES  800000
#define HID      128
#define LAYERS   4
#define OUTC     47
#define BN_EPS   1e-5f

typedef __attribute__((ext_vector_type(2))) float v2f;
typedef __attribute__((ext_vector_type(8))) float v8f;

// =====================================================================
// Degree accumulation: one thread per edge, native f32 atomics.
// =====================================================================
__global__ void k_degrees(const int* __restrict__ src, const int* __restrict__ dst,
                          float* __restrict__ out_deg, float* __restrict__ in_deg) {
    int e = blockIdx.x * blockDim.x + threadIdx.x;
    if (e >= N_EDGES) return;
    unsafeAtomicAdd(&out_deg[src[e]], 1.0f);
    unsafeAtomicAdd(&in_deg[dst[e]],  1.0f);
}

// norm = rsqrt(max(deg, 1))  -- covers out_deg and in_deg in one pass (2N elems)
__global__ void k_norm(const float* __restrict__ deg, float* __restrict__ nrm, int count) {
    int i = blockIdx.x * blockDim.x + threadIdx.x;
    if (i >= count) return;
    nrm[i] = rsqrtf(fmaxf(deg[i], 1.0f));
}

// =====================================================================
// FP32 WMMA GEMM:  out[N_NODES x NB] = diag(norm) * A[N_NODES x K] @ W[K x NB] (+bias)
//   - compile-time K, NB -> constant strides / trip counts / tile counts
//   - 16 rows per block (N_NODES = 3125 * 16, exact); 8 waves, 16-col tiles
//   - A tile staged in LDS (row stride K+4 dwords -> conflict-free operand reads)
//   - B loads: per-lane strided pointer (advance by 4*NB floats per k-step),
//     column clamped in-bounds + 0/1 mask -> no exec juggling in the hot loop
//
// Operand layout (CDNA5 ISA §7.12.2, 32-bit):
//   A 16x4: lanes 0-15 -> M=lane, VGPR{0,1} = K{k,k+1}; lanes 16-31 -> K{k+2,k+3}
//   B 4x16: lanes 0-15 -> N=lane, VGPR{0,1} = K{k,k+1}; lanes 16-31 -> K{k+2,k+3}
//   C/D 16x16: VGPR v -> M = v + 8*(lane>=16), N = lane&15
// =====================================================================
template<int K, int NB>
__global__ void k_gemm_wmma(const float* __restrict__ A,
                            const float* __restrict__ norm,   // may be null
                            const float* __restrict__ W,
                            const float* __restrict__ bias,   // may be null
                            float* __restrict__ out) {
    constexpr int LDSROW = K + 4;                 // dword stride, 4-bank skew
    __shared__ float smem[16 * LDSROW];
    const int row0 = blockIdx.x * 16;

    // ---- cooperative load of the (optionally pre-scaled) A tile ----
    constexpr int QPERROW = K >> 2;               // float4 chunks per row
    for (int i = threadIdx.x; i < 16 * QPERROW; i += blockDim.x) {
        const int r  = i / QPERROW;
        const int c4 = (i - r * QPERROW) << 2;
        const float4 v = *(const float4*)(A + (size_t)(row0 + r) * K + c4);
        const float  s = norm ? norm[row0 + r] : 1.0f;
        float* p = smem + r * LDSROW + c4;
        p[0] = v.x * s; p[1] = v.y * s; p[2] = v.z * s; p[3] = v.w * s;
    }
    __syncthreads();

    const int lane  = threadIdx.x & 31;
    const int wave  = threadIdx.x >> 5;
    const int lncol = lane & 15;
    const int khalf = (lane >> 4) << 1;           // 0 (lanes 0-15) or 2 (lanes 16-31)
    const float* arow = smem + lncol * LDSROW;    // A operand row: M = lane&15

    constexpr int  NTILES = (NB + 15) >> 4;
    constexpr bool FULL   = (NB & 15) == 0;       // no column guards needed

    for (int t = wave; t < NTILES; t += 8) {
        const int   n0    = t << 4;
        const int   col   = n0 + lncol;
        const bool  inb   = FULL ? true : (col < NB);
        const int   colc  = FULL ? col  : (inb ? col : (NB - 1));  // clamped, in-bounds
        const float wmask = FULL ? 1.0f : (inb ? 1.0f : 0.0f);

        const float* wp = W + (size_t)khalf * NB + colc;           // strided B pointer
        v8f acc = {};
        #pragma unroll 8
        for (int k = 0; k < K; k += 4) {
            const float2 av = *(const float2*)(arow + k + khalf);  // ds_load_b64
            v2f a; a.x = av.x; a.y = av.y;
            v2f b; b.x = wp[0] * wmask; b.y = wp[NB] * wmask;
            wp += 4 * NB;
            acc = __builtin_amdgcn_wmma_f32_16x16x4_f32(
                false, a, false, b, (short)0, acc, false, false);
        }
        if (inb) {
            const float bs    = bias ? bias[col] : 0.0f;
            const int   rbase = row0 + ((lane >> 4) << 3);
            #pragma unroll
            for (int v = 0; v < 8; ++v)
                out[(size_t)(rbase + v) * NB + col] = acc[v] + bs;
        }
    }
}

// =====================================================================
// Edge scatter-add: one wave per edge; lane handles 4 channels (float4 gather,
// 4 native f32 atomic adds).  agg[dst] += hw[src]
// =====================================================================
__global__ void k_scatter(const float* __restrict__ hw, const int* __restrict__ src,
                          const int* __restrict__ dst, float* __restrict__ agg) {
    const int wid = blockIdx.x * (blockDim.x >> 5) + (threadIdx.x >> 5);
    if (wid >= N_EDGES) return;
    const int lane = threadIdx.x & 31;
    const int s = src[wid];
    const int d = dst[wid];
    const float4 v = *(const float4*)(hw + (size_t)s * HID + (lane << 2));
    float* p = agg + (size_t)d * HID + (lane << 2);
    unsafeAtomicAdd(p + 0, v.x);
    unsafeAtomicAdd(p + 1, v.y);
    unsafeAtomicAdd(p + 2, v.z);
    unsafeAtomicAdd(p + 3, v.w);
}

// =====================================================================
// Fused layer epilogue: h = relu(BN(agg * norm_dst (+bias))) ; also writes
// the JK-concat slab (row stride HID*LAYERS).
// =====================================================================
__global__ void k_post(const float* __restrict__ agg, const float* __restrict__ norm_dst,
                       const float* __restrict__ bias,   // null except last layer
                       const float* __restrict__ gamma, const float* __restrict__ beta,
                       const float* __restrict__ mean,  const float* __restrict__ var,
                       float* __restrict__ hout, float* __restrict__ feats) {
    const size_t idx = (size_t)blockIdx.x * blockDim.x + threadIdx.x;
    if (idx >= (size_t)N_NODES * HID) return;
    const int    c = (int)(idx & (HID - 1));
    const size_t n = idx >> 7;
    float v = agg[idx] * norm_dst[n];
    if (bias) v += bias[c];
    v = (v - mean[c]) * (gamma[c] * rsqrtf(var[c] + BN_EPS)) + beta[c];
    v = fmaxf(v, 0.0f);
    hout[idx] = v;
    feats[n * (size_t)(HID * LAYERS) + c] = v;
}

// Classifier BN + ReLU (128 channels)
__global__ void k_bnrelu(const float* __restrict__ x,
                         const float* __restrict__ gamma, const float* __restrict__ beta,
                         const float* __restrict__ mean,  const float* __restrict__ var,
                         float* __restrict__ y) {
    const size_t idx = (size_t)blockIdx.x * blockDim.x + threadIdx.x;
    if (idx >= (size_t)N_NODES * HID) return;
    const int c = (int)(idx & (HID - 1));
    float v = (x[idx] - mean[c]) * (gamma[c] * rsqrtf(var[c] + BN_EPS)) + beta[c];
    y[idx] = fmaxf(v, 0.0f);
}

// =====================================================================
extern "C" void kernel_launch(void* const* d_in, const int* in_sizes, int n_in,
                              void* d_out, int out_size, void* d_ws, size_t ws_size,
                              hipStream_t stream) {
    const float* feat      = (const float*)d_in[0];
    const int*   src       = (const int*)  d_in[1];
    const int*   dst       = (const int*)  d_in[2];
    const float* enc_W     = (const float*)d_in[3];   // [L,128,128] row-major
    const float* enc_bias  = (const float*)d_in[4];
    const float* bn_gamma  = (const float*)d_in[5];
    const float* bn_beta   = (const float*)d_in[6];
    const float* bn_mean   = (const float*)d_in[7];
    const float* bn_var    = (const float*)d_in[8];
    const float* cls_W0    = (const float*)d_in[9];   // [512,128]
    const float* cls_b0    = (const float*)d_in[10];
    const float* cls_g0    = (const float*)d_in[11];
    const float* cls_be0   = (const float*)d_in[12];
    const float* cls_m0    = (const float*)d_in[13];
    const float* cls_v0    = (const float*)d_in[14];
    const float* cls_W1    = (const float*)d_in[15];  // [128,47]
    const float* cls_b1    = (const float*)d_in[16];

    const size_t N = N_NODES;
    float* ws   = (float*)d_ws;
    float* deg  = ws;                 // 2N  (out_deg | in_deg)
    float* nrm  = ws + 2 * N;         // 2N  (norm_src | norm_dst)
    float* hbuf = ws + 4 * N;         // N*128
    float* hw   = hbuf + (size_t)HID * N;   // N*128
    float* agg  = hw   + (size_t)HID * N;   // N*128
    float* feats= agg  + (size_t)HID * N;   // N*512

    // degrees -> norms
    hipMemsetAsync(deg, 0, 2 * N * sizeof(float), stream);
    k_degrees<<<(N_EDGES + 255) / 256, 256, 0, stream>>>(src, dst, deg, deg + N);
    k_norm<<<(int)((2 * N + 255) / 256), 256, 0, stream>>>(deg, nrm, (int)(2 * N));
    const float* norm_src = nrm;
    const float* norm_dst = nrm + N;

    const int gblocks = N_NODES / 16;                 // 3125, exact
    const int eblocks = (N_EDGES + 7) / 8;            // 8 waves / block
    const int pblocks = (int)(((size_t)N * HID + 255) / 256);

    for (int l = 0; l < LAYERS; ++l) {
        const float* Ain = (l == 0) ? feat : hbuf;
        // hw = diag(norm_src) * h @ enc_W[l]
        k_gemm_wmma<128, 128><<<gblocks, 256, 0, stream>>>(
            Ain, norm_src, enc_W + (size_t)l * HID * HID, nullptr, hw);
        // agg = segment_sum(hw[src], dst)
        hipMemsetAsync(agg, 0, (size_t)HID * N * sizeof(float), stream);
        k_scatter<<<eblocks, 256, 0, stream>>>(hw, src, dst, agg);
        // h = relu(BN(agg * norm_dst (+bias on last))); also fill JK slab
        k_post<<<pblocks, 256, 0, stream>>>(
            agg, norm_dst, (l == LAYERS - 1) ? enc_bias : nullptr,
            bn_gamma + l * HID, bn_beta + l * HID, bn_mean + l * HID, bn_var + l * HID,
            hbuf, feats + l * HID);
    }

    // classifier layer 0: feats[N,512] @ W0 + b0 -> BN -> ReLU
    k_gemm_wmma<512, 128><<<gblocks, 256, 0, stream>>>(feats, nullptr, cls_W0, cls_b0, hw);
    k_bnrelu<<<pblocks, 256, 0, stream>>>(hw, cls_g0, cls_be0, cls_m0, cls_v0, hbuf);

    // classifier layer 1: hbuf[N,128] @ W1 + b1 -> d_out[N,47]
    k_gemm_wmma<128, OUTC><<<gblocks, 256, 0, stream>>>(hbuf, nullptr, cls_W1, cls_b1,
                                                        (float*)d_out);
}